// RoIAlignWrapper_3874060501363
// MI455X (gfx1250) — compile-verified
//
#include <hip/hip_runtime.h>
#include <hip/hip_bf16.h>

#define ROI_B 8
#define ROI_N 64
#define ROI_C 256
#define ROI_H 128
#define ROI_W 128
#define OUT_HW 36
#define TILE_FLOATS 12288            // 48KB per buffer
#define TILE_BYTES (TILE_FLOATS * 4)

typedef unsigned int uint32x4 __attribute__((ext_vector_type(4)));
typedef int int32x8 __attribute__((ext_vector_type(8)));
typedef int int32x4 __attribute__((ext_vector_type(4)));

__device__ __forceinline__ unsigned rfl_u(unsigned v) {
  return (unsigned)__builtin_amdgcn_readfirstlane((int)v);
}

// Exact torchvision bilinear_interpolate 1-D semantics (matches reference _interp_1d)
__device__ __forceinline__ void interp1d(float coord, int size, int& lo, int& hi,
                                         float& w0, float& w1) {
  bool valid = (coord >= -1.0f) && (coord <= (float)size);
  float c = fmaxf(coord, 0.0f);
  int low0 = (int)floorf(c);
  bool at_edge = (low0 >= size - 1);
  lo = at_edge ? (size - 1) : low0;
  hi = at_edge ? (size - 1) : (low0 + 1);
  float frac = at_edge ? 0.0f : (c - (float)low0);
  w0 = valid ? (1.0f - frac) : 0.0f;
  w1 = valid ? frac : 0.0f;
}

// Build a 3-D TDM descriptor (x=cols, y=rows, z=channels) and fire it.
// Must be executed by exactly one wave; EXEC is ignored by TDM.
__device__ __forceinline__ void tdm_load_tile(const float* xin, unsigned lds_byte_addr,
                                              int bi, int c0, int y0, int x0,
                                              int bw, int bh, int ccc) {
  const unsigned long long ga =
      (unsigned long long)(const void*)xin +
      4ull * ((((unsigned long long)((unsigned)bi * ROI_C + (unsigned)c0)) * ROI_H +
               (unsigned long long)(unsigned)y0) * ROI_W +
              (unsigned long long)(unsigned)x0);
  const unsigned galo = rfl_u((unsigned)ga);
  const unsigned gahi = rfl_u((unsigned)(ga >> 32));
  const unsigned td0 = 128u - (unsigned)x0;  // remaining tensor extent dim0 (x)
  const unsigned td1 = 128u - (unsigned)y0;  // remaining extent dim1 (y)
  const unsigned td2 = 256u - (unsigned)c0;  // remaining extent dim2 (channels)
  // --- D# group 0: count=1, lds_addr, 57b global_addr, type=2 ---
  uint32x4 g0;
  g0.x = 1u;
  g0.y = rfl_u(lds_byte_addr);
  g0.z = galo;
  g0.w = (gahi & 0x01FFFFFFu) | (2u << 30);
  // --- D# group 1: data_size=4B; dims/strides; tile bw x bh x ccc ---
  int32x8 g1;
  g1.s0 = (int)(2u << 16);                                               // data_size=4B
  g1.s1 = (int)rfl_u((td0 & 0xFFFFu) << 16);                             // tensor_dim0[15:0]
  g1.s2 = (int)rfl_u(((td0 >> 16) & 0xFFFFu) | ((td1 & 0xFFFFu) << 16));
  g1.s3 = (int)rfl_u(((td1 >> 16) & 0xFFFFu) | (((unsigned)bw & 0xFFFFu) << 16));   // tile_dim0
  g1.s4 = (int)rfl_u(((unsigned)bh & 0xFFFFu) | (((unsigned)ccc & 0xFFFFu) << 16)); // tile_dim1|2
  g1.s5 = 128;                  // tensor_dim0_stride = W (row stride)
  g1.s6 = (int)(16384u << 16);  // tensor_dim1_stride = H*W (channel stride) in bits [223:208]
  g1.s7 = 0;                    // tensor_dim1_stride[47:16] = 0
  // --- D# group 2: tensor_dim2 = remaining channels; 3-D tile ---
  int32x4 g2;
  g2.x = (int)rfl_u(td2); g2.y = 0; g2.z = 0; g2.w = 0;
  int32x4 g3;
  g3.x = 0; g3.y = 0; g3.z = 0; g3.w = 0;
#if defined(__clang_major__) && (__clang_major__ >= 23)
  int32x8 g4 = {0, 0, 0, 0, 0, 0, 0, 0};
  __builtin_amdgcn_tensor_load_to_lds(g0, g1, g2, g3, g4, 0);
#else
  __builtin_amdgcn_tensor_load_to_lds(g0, g1, g2, g3, 0);
#endif
}

__global__ __launch_bounds__(288, 1) void roi_align_tdm_kernel(
    const float* __restrict__ xin, const float* __restrict__ boxes,
    float* __restrict__ out) {
  __shared__ float tile[2][TILE_FLOATS];       // double-buffered TDM tiles (96KB)
  __shared__ float sW[OUT_HW * 16];            // per-pixel 16 combined weights
  __shared__ int sLoc[OUT_HW * 16];            // tile-local offsets
  __shared__ int sGlb[OUT_HW * 16];            // absolute y*W+x offsets (fallback)
  __shared__ int sYL[12], sYH[12], sXL[12], sXH[12];
  __shared__ float sWY0[12], sWY1[12], sWX0[12], sWX1[12];
  __shared__ int sMeta[6];                     // y0, x0, bw, bh, area, cc

  const int tid = threadIdx.x;
  const int bi = blockIdx.x >> 6;
  const int n = blockIdx.x & 63;

  const float* bx = boxes + (size_t)(bi * ROI_N + n) * 4;
  const float x1 = bx[0] - 0.5f;  // aligned=True half-pixel offset
  const float y1 = bx[1] - 0.5f;
  const float x2 = bx[2] - 0.5f;
  const float y2 = bx[3] - 0.5f;
  const float bin_h = (y2 - y1) / 6.0f;
  const float bin_w = (x2 - x1) / 6.0f;

  // ---- Stage 1a: 12 y-sample + 12 x-sample 1-D coefficients ----
  if (tid < 24) {
    const int j = (tid < 12) ? tid : (tid - 12);
    const float pos = (float)(j >> 1) + (float)(j & 1) * 0.5f + 0.25f;
    if (tid < 12) {
      interp1d(y1 + pos * bin_h, ROI_H, sYL[j], sYH[j], sWY0[j], sWY1[j]);
    } else {
      interp1d(x1 + pos * bin_w, ROI_W, sXL[j], sXH[j], sWX0[j], sWX1[j]);
    }
  }
  __syncthreads();

  // ---- Stage 1b: bounding tile + channels-per-chunk ----
  if (tid == 0) {
    int y0 = ROI_H - 1, ym = 0, x0 = ROI_W - 1, xm = 0;
    for (int j = 0; j < 12; ++j) {
      y0 = (sYL[j] < y0) ? sYL[j] : y0;
      ym = (sYH[j] > ym) ? sYH[j] : ym;
      x0 = (sXL[j] < x0) ? sXL[j] : x0;
      xm = (sXH[j] > xm) ? sXH[j] : xm;
    }
    const int bw = xm - x0 + 1;
    const int bh = ym - y0 + 1;
    const int area = bw * bh;
    int cc = 0;
    if (area <= TILE_FLOATS) {
      cc = TILE_FLOATS / area;
      cc = (cc > 8) ? 8 : cc;
    }
    sMeta[0] = y0; sMeta[1] = x0; sMeta[2] = bw; sMeta[3] = bh;
    sMeta[4] = area; sMeta[5] = cc;
  }
  __syncthreads();

  // ---- Stage 1c: per-pixel 16-term (weight, offset) tables ----
  if (tid < OUT_HW) {
    const int oh = tid / 6, ow = tid - oh * 6;
    const int ty0 = sMeta[0], tx0 = sMeta[1], tbw = sMeta[2];
    for (int a = 0; a < 2; ++a) {
      const int sy = 2 * oh + a;
      const int yl = sYL[sy], yh = sYH[sy];
      const float wy0 = sWY0[sy], wy1 = sWY1[sy];
      for (int bb = 0; bb < 2; ++bb) {
        const int sx = 2 * ow + bb;
        const int xl = sXL[sx], xh = sXH[sx];
        const float wx0 = sWX0[sx], wx1 = sWX1[sx];
        const int base = tid * 16 + (a * 2 + bb) * 4;
        sW[base + 0] = 0.25f * wy0 * wx0;  // 0.25 = 2x2 sample mean
        sW[base + 1] = 0.25f * wy0 * wx1;
        sW[base + 2] = 0.25f * wy1 * wx0;
        sW[base + 3] = 0.25f * wy1 * wx1;
        sGlb[base + 0] = yl * ROI_W + xl;
        sGlb[base + 1] = yl * ROI_W + xh;
        sGlb[base + 2] = yh * ROI_W + xl;
        sGlb[base + 3] = yh * ROI_W + xh;
        sLoc[base + 0] = (yl - ty0) * tbw + (xl - tx0);
        sLoc[base + 1] = (yl - ty0) * tbw + (xh - tx0);
        sLoc[base + 2] = (yh - ty0) * tbw + (xl - tx0);
        sLoc[base + 3] = (yh - ty0) * tbw + (xh - tx0);
      }
    }
  }
  __syncthreads();

  const int p = tid % 36;       // output pixel
  const int cslot = tid / 36;   // channel slot 0..7 (288 = 8*36)
  float wr[16]; int lr[16]; int gr[16];
  #pragma unroll
  for (int t = 0; t < 16; ++t) {
    wr[t] = sW[p * 16 + t];
    lr[t] = sLoc[p * 16 + t];
    gr[t] = sGlb[p * 16 + t];
  }
  const int y0 = sMeta[0], x0 = sMeta[1], bw = sMeta[2], bh = sMeta[3];
  const int area = sMeta[4], cc = sMeta[5];
  float* const outb = out + (size_t)(bi * ROI_N + n) * (ROI_C * OUT_HW);

  if (cc > 0) {
    // ---- Pipelined TDM path: DMA chunk k+1 overlaps compute of chunk k ----
    const unsigned lds0 = (unsigned)(unsigned long long)(const void*)&tile[0][0];
    const int nchunks = (ROI_C + cc - 1) / cc;
    if (tid < 32) {
      const int ccc0 = (cc < ROI_C) ? cc : ROI_C;
      tdm_load_tile(xin, lds0, bi, 0, y0, x0, bw, bh, ccc0);  // prologue: chunk 0
    }
    for (int k = 0; k < nchunks; ++k) {
      const int c0 = k * cc;
      const int ccc = (cc < ROI_C - c0) ? cc : (ROI_C - c0);
      if (tid < 32) {
        const int c1 = c0 + cc;
        if (c1 < ROI_C) {  // issue chunk k+1 into the other buffer
          const int ccn = (cc < ROI_C - c1) ? cc : (ROI_C - c1);
          tdm_load_tile(xin, lds0 + (unsigned)((k + 1) & 1) * TILE_BYTES,
                        bi, c1, y0, x0, bw, bh, ccn);
          __builtin_amdgcn_s_wait_tensorcnt(1);  // chunk k done; k+1 in flight
        } else {
          __builtin_amdgcn_s_wait_tensorcnt(0);  // last chunk: drain
        }
      }
      __syncthreads();  // tile[k&1] now valid for all waves
      if (cslot < ccc) {
        const float* tb = &tile[k & 1][0] + (size_t)((unsigned)cslot * (unsigned)area);
        float acc = 0.0f;
        #pragma unroll
        for (int t = 0; t < 16; ++t) acc = fmaf(wr[t], tb[lr[t]], acc);
        outb[(size_t)(c0 + cslot) * OUT_HW + p] = acc;
      }
      __syncthreads();  // protect tile[k&1] before it is re-issued at chunk k+2
    }
  } else {
    // ---- Fallback for oversized boxes: direct global gather (L2-resident) ----
    const float* const plane0 = xin + ((size_t)((unsigned)bi * ROI_C) << 14);
    for (int c0 = 0; c0 < ROI_C; c0 += 8) {
      const int c = c0 + cslot;
      const float* src = plane0 + ((size_t)(unsigned)c << 14);
      float acc = 0.0f;
      #pragma unroll
      for (int t = 0; t < 16; ++t) acc = fmaf(wr[t], src[gr[t]], acc);
      outb[(size_t)c * OUT_HW + p] = acc;
    }
  }
}

extern "C" void kernel_launch(void* const* d_in, const int* in_sizes, int n_in,
                              void* d_out, int out_size, void* d_ws, size_t ws_size,
                              hipStream_t stream) {
  (void)in_sizes; (void)n_in; (void)out_size; (void)d_ws; (void)ws_size;
  const float* x = (const float*)d_in[0];      // [8,256,128,128]
  const float* boxes = (const float*)d_in[1];  // [8,64,4]
  float* out = (float*)d_out;                  // [8,64,256,6,6]
  roi_align_tdm_kernel<<<dim3(ROI_B * ROI_N), dim3(288), 0, stream>>>(x, boxes, out);
}